// QModel_71021579206864
// MI455X (gfx1250) — compile-verified
//
#include <hip/hip_runtime.h>

// ---------------------------------------------------------------------------
// Fused quantum-LSTM forward for MI455X (gfx1250, wave32).
//
// One kernel, one pass. Thread = one batch row: h[16], c[16], and the 16
// complex amplitudes of the 4-qubit simulator live in VGPRs (fully unrolled).
// The two tiny per-step GEMMs ([B,17]x[17,4] and [B,4]x[4,16]) are done with
// V_WMMA_F32_16X16X4_F32 on 16-row batch tiles (2 tiles per wave32), with LDS
// used to move between the per-thread layout and the ISA WMMA VGPR layouts.
// ---------------------------------------------------------------------------

typedef __attribute__((ext_vector_type(2))) float v2f;
typedef __attribute__((ext_vector_type(8))) float v8f;

#define INV_SQRT2 0.70710678118654752440f

// CNOT on the 16-amplitude real state: amp_new[i] = amp[i ^ TM] when (i & CM).
template <int CM, int TM>
__device__ __forceinline__ void cnot_swap(float* a) {
#pragma unroll
  for (int i = 0; i < 16; ++i) {
    if ((i & CM) && !(i & TM)) {
      float t = a[i | TM];
      a[i | TM] = a[i];
      a[i] = t;
    }
  }
}

// RX(theta) on qubit with basis mask M: U = [[cr, -i*s], [-i*s, cr]].
template <int M>
__device__ __forceinline__ void rx_gate(float* re, float* im, float cr, float s) {
#pragma unroll
  for (int i = 0; i < 16; ++i) {
    if (!(i & M)) {
      const int j = i | M;
      const float r0 = re[i], i0 = im[i], r1 = re[j], i1 = im[j];
      re[i] = cr * r0 + s * i1;
      im[i] = cr * i0 - s * r1;
      re[j] = s * i0 + cr * r1;
      im[j] = -s * r0 + cr * i1;
    }
  }
}

__global__ __launch_bounds__(256) void qlstm_kernel(
    const float* __restrict__ x,   // [B,4,1]
    const float* __restrict__ h0,  // [1,B,16]
    const float* __restrict__ c0,  // [1,B,16]
    const float* __restrict__ w,   // [4]
    const float* __restrict__ W1,  // [4,17]
    const float* __restrict__ b1,  // [4]
    const float* __restrict__ W2,  // [16,4]
    const float* __restrict__ b2,  // [16]
    float* __restrict__ out,       // [B,16]
    int B) {
  // Shared staging. sbuf doubles as the h-publish buffer (read as WMMA1 A) and
  // the z result buffer (written from WMMA2 D). fbuf doubles as y then e.
  // Row pitch 17 floats -> column reads (fixed k, m = lane) hit distinct banks.
  __shared__ float sbuf[256][17];
  __shared__ float fbuf[256][4];

  const int tid = threadIdx.x;
  const int b = blockIdx.x * 256 + tid;
  const int lane = tid & 31;
  const int wv = tid >> 5;       // wave id in block (8 waves)
  const int hi = lane >> 4;      // half-wave select
  const int n16 = lane & 15;     // N / M index within a 16-wide tile

  // ---- per-thread recurrent state ----
  float h[16], c[16];
#pragma unroll
  for (int j = 0; j < 16; ++j) {
    h[j] = h0[b * 16 + j];
    c[j] = c0[b * 16 + j];
  }
  const float4 xv = ((const float4*)x)[b];  // x[b, 0..3, 0]

  // ---- uniform small parameters ----
  float wc[4], ws[4], b1r[4], w1x[4];
#pragma unroll
  for (int i = 0; i < 4; ++i) {
    const float th = 0.5f * w[i];
    wc[i] = cosf(th);
    ws[i] = sinf(th);
    b1r[i] = b1[i];
    w1x[i] = W1[i * 17 + 16];  // x column of W1
  }
  const float b2l = b2[n16];

  // ---- WMMA B fragments (loop-invariant) ----
  // B layout (4x16 f32): VGPR j -> row K = j + 2*hi, col N = lane&15.
  // Matmul1: B[k][n] = W1[n][k] for n<4 (h part of W1), 0-padded to N=16.
  v2f bw1[4];
#pragma unroll
  for (int kc = 0; kc < 4; ++kc) {
    float bx = 0.0f, by = 0.0f;
    if (n16 < 4) {
      bx = W1[n16 * 17 + 4 * kc + 2 * hi + 0];
      by = W1[n16 * 17 + 4 * kc + 2 * hi + 1];
    }
    v2f t = {bx, by};
    bw1[kc] = t;
  }
  // Matmul2: B[k][n] = W2[n][k], K=4, N=16.
  v2f bw2 = {W2[n16 * 4 + 2 * hi + 0], W2[n16 * 4 + 2 * hi + 1]};

  for (int t = 0; t < 4; ++t) {
    // (1) publish h for this step's WMMA1 A operand
#pragma unroll
    for (int j = 0; j < 16; ++j) sbuf[tid][j] = h[j];
    __syncthreads();

    // (2) y_tile = h_tile @ W1h^T via 4 chained K=4 WMMAs; 2 tiles per wave.
#pragma unroll
    for (int tile = 0; tile < 2; ++tile) {
      const int tb = wv * 32 + tile * 16;
      v8f acc = {0.f, 0.f, 0.f, 0.f, 0.f, 0.f, 0.f, 0.f};
#pragma unroll
      for (int kc = 0; kc < 4; ++kc) {
        // A layout (16x4 f32): lane&15 = M, VGPR j -> K = 4*kc + j + 2*hi.
        const int kb = 4 * kc + 2 * hi;
        v2f a = {sbuf[tb + n16][kb], sbuf[tb + n16][kb + 1]};
        acc = __builtin_amdgcn_wmma_f32_16x16x4_f32(
            false, a, false, bw1[kc], (short)0, acc, false, false);
      }
      // D layout: lane&15 = N (y component, valid for N<4), VGPR v -> M=v+8*hi.
      if (n16 < 4) {
#pragma unroll
        for (int v = 0; v < 8; ++v) fbuf[tb + v + 8 * hi][n16] = acc[v];
      }
    }
    __syncthreads();

    // (3) per-thread 4-qubit circuit on y = h@W1h^T + x*W1x + b1
    const float xt = (t == 0) ? xv.x : (t == 1) ? xv.y : (t == 2) ? xv.z : xv.w;
    float e[4];
    {
      float q[4][2];
#pragma unroll
      for (int i = 0; i < 4; ++i) {
        const float y = fbuf[tid][i] + b1r[i] + xt * w1x[i];
        const float a = atanf(y);
        const float cc = cosf(0.5f * a);
        const float ss = sinf(0.5f * a);
        q[i][0] = (cc - ss) * INV_SQRT2;  // |0> amplitude after H, RY(a)
        q[i][1] = (cc + ss) * INV_SQRT2;  // |1> amplitude
      }
      // outer product -> 16 real amplitudes; qubit q <-> bit (3-q)
      float amp[16];
#pragma unroll
      for (int i = 0; i < 16; ++i)
        amp[i] = q[0][(i >> 3) & 1] * q[1][(i >> 2) & 1] *
                 q[2][(i >> 1) & 1] * q[3][i & 1];
      cnot_swap<8, 4>(amp);  // CNOT(0,1)
      cnot_swap<4, 2>(amp);  // CNOT(1,2)
      cnot_swap<2, 1>(amp);  // CNOT(2,3)
      cnot_swap<1, 8>(amp);  // CNOT(3,0)
      float re[16], im[16];
#pragma unroll
      for (int i = 0; i < 16; ++i) { re[i] = amp[i]; im[i] = 0.0f; }
      rx_gate<8>(re, im, wc[0], ws[0]);
      rx_gate<4>(re, im, wc[1], ws[1]);
      rx_gate<2>(re, im, wc[2], ws[2]);
      rx_gate<1>(re, im, wc[3], ws[3]);
      float p[16];
#pragma unroll
      for (int i = 0; i < 16; ++i) p[i] = re[i] * re[i] + im[i] * im[i];
#pragma unroll
      for (int i = 0; i < 4; ++i) {
        const int m = 8 >> i;
        float s = 0.0f;
#pragma unroll
        for (int idx = 0; idx < 16; ++idx) s += (idx & m) ? -p[idx] : p[idx];
        e[i] = s;
      }
    }
#pragma unroll
    for (int i = 0; i < 4; ++i) fbuf[tid][i] = e[i];
    __syncthreads();

    // (4) z_tile = sigmoid(e_tile @ W2^T + b2) via one K=4 WMMA per tile.
#pragma unroll
    for (int tile = 0; tile < 2; ++tile) {
      const int tb = wv * 32 + tile * 16;
      v2f a = {fbuf[tb + n16][2 * hi], fbuf[tb + n16][2 * hi + 1]};
      v8f acc = {0.f, 0.f, 0.f, 0.f, 0.f, 0.f, 0.f, 0.f};
      acc = __builtin_amdgcn_wmma_f32_16x16x4_f32(
          false, a, false, bw2, (short)0, acc, false, false);
#pragma unroll
      for (int v = 0; v < 8; ++v) {
        const float zp = acc[v] + b2l;
        sbuf[tb + v + 8 * hi][n16] = 1.0f / (1.0f + expf(-zp));
      }
    }
    __syncthreads();

    // (5) cell / hidden update (source has f==i==g==o==z)
#pragma unroll
    for (int j = 0; j < 16; ++j) {
      const float z = sbuf[tid][j];
      c[j] = z * c[j] + z * z;
      h[j] = z * tanhf(c[j]);
    }
    // Row ownership makes the loop-back write to sbuf[tid] race-free.
  }

  float4* o = (float4*)(out + b * 16);
#pragma unroll
  for (int j = 0; j < 4; ++j)
    o[j] = make_float4(h[4 * j + 0], h[4 * j + 1], h[4 * j + 2], h[4 * j + 3]);
}

extern "C" void kernel_launch(void* const* d_in, const int* in_sizes, int n_in,
                              void* d_out, int out_size, void* d_ws,
                              size_t ws_size, hipStream_t stream) {
  const float* x  = (const float*)d_in[0];
  const float* h0 = (const float*)d_in[1];
  const float* c0 = (const float*)d_in[2];
  const float* w  = (const float*)d_in[3];
  const float* W1 = (const float*)d_in[4];
  const float* b1 = (const float*)d_in[5];
  const float* W2 = (const float*)d_in[6];
  const float* b2 = (const float*)d_in[7];
  float* out = (float*)d_out;

  const int B = in_sizes[0] / 4;   // x is [B,4,1]
  const int nblk = B / 256;        // B = 65536 -> 256 blocks of 256 threads
  qlstm_kernel<<<dim3(nblk), dim3(256), 0, stream>>>(x, h0, c0, w, W1, b1, W2,
                                                     b2, out, B);
}